// P3dPose_33560874451540
// MI455X (gfx1250) — compile-verified
//
#include <hip/hip_runtime.h>
#include <cstdint>
#include <cstddef>

#define NB 26
#define TPB 128                       // 4 waves of 32
#define ELEM_FLOATS (NB * 3)          // 78 floats / element
#define ELEM_BYTES  (ELEM_FLOATS * 4) // 312 B
#define WAVE_BYTES  (32 * ELEM_BYTES) // 9984 B staged per wave
#define BLOCK_BYTES (TPB * ELEM_BYTES)
#define CHUNKS128   (WAVE_BYTES / (32 * 16)) // 19 full b128 wave-iterations
// tail: WAVE_BYTES - 19*512 = 256 B -> one b128 issue by lanes 0..15

__device__ __constant__ int kParents[NB] = {
    -1, 0, 1, 2, 3, 4, 5, 4, 7, 8, 9, 4, 11, 12, 13,
    1, 15, 16, 17, 1, 19, 20, 21, 6, 6, 6};

// ---------------------------------------------------------------------------
// Setup: per-bone constant table, 8 floats per bone:
//   [0..4] = the 5 entries of R_ref euler-extraction reads
//            (cz*cy, sz*cy, -sy, cy*sx, cy*cx)
//   [5..7] = rel_loc
// ---------------------------------------------------------------------------
__global__ void fk_setup(const float* __restrict__ rel_rot_ref,
                         const float* __restrict__ rel_loc,
                         float* __restrict__ cst) {
  int i = threadIdx.x;
  if (i < NB) {
    float sz, cz, sy, cy, sx, cx;
    __sincosf(rel_rot_ref[i * 3 + 0], &sz, &cz);
    __sincosf(rel_rot_ref[i * 3 + 1], &sy, &cy);
    __sincosf(rel_rot_ref[i * 3 + 2], &sx, &cx);
    cst[i * 8 + 0] = cz * cy;
    cst[i * 8 + 1] = sz * cy;
    cst[i * 8 + 2] = -sy;
    cst[i * 8 + 3] = cy * sx;
    cst[i * 8 + 4] = cy * cx;
    cst[i * 8 + 5] = rel_loc[i * 3 + 0];
    cst[i * 8 + 6] = rel_loc[i * 3 + 1];
    cst[i * 8 + 7] = rel_loc[i * 3 + 2];
  }
}

// ---------------------------------------------------------------------------
// Main FK kernel: async-stage 312B/thread into LDS (coalesced b128), run the
// fully-unrolled 26-bone chain in registers, overwrite LDS in place with
// abs_loc, async-store back out.
// ---------------------------------------------------------------------------
__global__ void __launch_bounds__(TPB) fk_kernel(
    const float* __restrict__ x, const float* __restrict__ cst,
    float* __restrict__ out) {
  __shared__ __align__(16) float smem[TPB * ELEM_FLOATS];

  const unsigned tid  = threadIdx.x;
  const unsigned lane = tid & 31u;
  const unsigned wave = tid >> 5;
  const unsigned ldsBase =
      (unsigned)(uintptr_t)(&smem[0]) + wave * WAVE_BYTES + lane * 16u;

  // ---- coalesced async global -> LDS (b128, tracked by ASYNCcnt) ----
  {
    const char* g = (const char*)x + (size_t)blockIdx.x * BLOCK_BYTES +
                    (size_t)wave * WAVE_BYTES + lane * 16u;
    unsigned l = ldsBase;
#pragma unroll
    for (int k = 0; k < CHUNKS128; ++k) {
      asm volatile("global_load_async_to_lds_b128 %0, %1, off"
                   :
                   : "v"(l), "v"(g)
                   : "memory");
      g += 512;
      l += 512;
    }
    if (lane < 16u) {  // 256 B tail of the wave's region
      asm volatile("global_load_async_to_lds_b128 %0, %1, off"
                   :
                   : "v"(l), "v"(g)
                   : "memory");
    }
    asm volatile("s_wait_asynccnt 0x0" ::: "memory");
  }

  float* slot = &smem[tid * ELEM_FLOATS];

  float A[NB][9];
  float B[NB][3];

#pragma unroll
  for (int i = 0; i < NB; ++i) {
    const int p = kParents[i];

    float P00, P01, P02, P10, P11, P12, P20, P21, P22, b0, b1, b2;
    if (p < 0) {
      P00 = 1.f; P01 = 0.f; P02 = 0.f;
      P10 = 0.f; P11 = 1.f; P12 = 0.f;
      P20 = 0.f; P21 = 0.f; P22 = 1.f;
      b0 = 0.f; b1 = 0.f; b2 = 0.f;
    } else {
      P00 = A[p][0]; P01 = A[p][1]; P02 = A[p][2];
      P10 = A[p][3]; P11 = A[p][4]; P12 = A[p][5];
      P20 = A[p][6]; P21 = A[p][7]; P22 = A[p][8];
      b0 = B[p][0]; b1 = B[p][1]; b2 = B[p][2];
    }

    const float* c = cst + i * 8;
    const float l0 = c[5], l1 = c[6], l2 = c[7];

    // abs_loc[i] = li^T * Ap + bp   (only needs parent state)
    const float o0 = fmaf(l0, P00, fmaf(l1, P10, fmaf(l2, P20, b0)));
    const float o1 = fmaf(l0, P01, fmaf(l1, P11, fmaf(l2, P21, b1)));
    const float o2 = fmaf(l0, P02, fmaf(l1, P12, fmaf(l2, P22, b2)));

    // consume euler, then overwrite slot in place with the output
    const float e0 = slot[i * 3 + 0];
    const float e1 = slot[i * 3 + 1];
    const float e2 = slot[i * 3 + 2];
    slot[i * 3 + 0] = o0;
    slot[i * 3 + 1] = o1;
    slot[i * 3 + 2] = o2;

    float sz, cz, sy, cy, sx, cx;
    __sincosf(e0, &sz, &cz);
    __sincosf(e1, &sy, &cy);
    __sincosf(e2, &sx, &cx);

    // Hadamard(R_ref, R_chg): only the 5 entries euler extraction reads.
    const float m00 = c[0] * (cz * cy);
    const float m10 = c[1] * (sz * cy);
    const float m20 = c[2] * (-sy);
    const float m21 = c[3] * (cy * sx);
    const float m22 = c[4] * (cy * cx);

    // matrix->euler->matrix collapses: sin(atan2(a,b)) = a*rsqrt(a^2+b^2),
    // sin(asin(t)) = t, cos(asin(t)) = sqrt(1-t^2). No atan2/asin needed.
    const float n1  = fmaf(m00, m00, m10 * m10);
    const float rn1 = (n1 > 0.f) ? rsqrtf(n1) : 0.f;
    const float SZ  = m10 * rn1;
    const float CZ  = (n1 > 0.f) ? m00 * rn1 : 1.f;
    const float SY  = -m20;
    const float CY  = sqrtf(fmaxf(fmaf(-m20, m20, 1.f), 0.f));
    const float n2  = fmaf(m21, m21, m22 * m22);
    const float rn2 = (n2 > 0.f) ? rsqrtf(n2) : 0.f;
    const float SX  = m21 * rn2;
    const float CX  = (n2 > 0.f) ? m22 * rn2 : 1.f;

    // rebuild full R_i (zyx)
    const float R00 = CZ * CY;
    const float R01 = CZ * SY * SX - SZ * CX;
    const float R02 = CZ * SY * CX + SZ * SX;
    const float R10 = SZ * CY;
    const float R11 = SZ * SY * SX + CZ * CX;
    const float R12 = SZ * SY * CX - CZ * SX;
    const float R20 = -SY;
    const float R21 = CY * SX;
    const float R22 = CY * CX;

    // A[i] = Ap @ R_i
    A[i][0] = fmaf(P00, R00, fmaf(P01, R10, P02 * R20));
    A[i][1] = fmaf(P00, R01, fmaf(P01, R11, P02 * R21));
    A[i][2] = fmaf(P00, R02, fmaf(P01, R12, P02 * R22));
    A[i][3] = fmaf(P10, R00, fmaf(P11, R10, P12 * R20));
    A[i][4] = fmaf(P10, R01, fmaf(P11, R11, P12 * R21));
    A[i][5] = fmaf(P10, R02, fmaf(P11, R12, P12 * R22));
    A[i][6] = fmaf(P20, R00, fmaf(P21, R10, P22 * R20));
    A[i][7] = fmaf(P20, R01, fmaf(P21, R11, P22 * R21));
    A[i][8] = fmaf(P20, R02, fmaf(P21, R12, P22 * R22));

    // B[i] = (bp + li)^T @ R_i
    const float t0 = b0 + l0, t1 = b1 + l1, t2 = b2 + l2;
    B[i][0] = fmaf(t0, R00, fmaf(t1, R10, t2 * R20));
    B[i][1] = fmaf(t0, R01, fmaf(t1, R11, t2 * R21));
    B[i][2] = fmaf(t0, R02, fmaf(t1, R12, t2 * R22));
  }

  // ---- flush own-wave DS stores, then coalesced async LDS -> global ----
  asm volatile("s_wait_dscnt 0x0" ::: "memory");
  {
    char* g = (char*)out + (size_t)blockIdx.x * BLOCK_BYTES +
              (size_t)wave * WAVE_BYTES + lane * 16u;
    unsigned l = ldsBase;
#pragma unroll
    for (int k = 0; k < CHUNKS128; ++k) {
      asm volatile("global_store_async_from_lds_b128 %0, %1, off"
                   :
                   : "v"(g), "v"(l)
                   : "memory");
      g += 512;
      l += 512;
    }
    if (lane < 16u) {  // 256 B tail
      asm volatile("global_store_async_from_lds_b128 %0, %1, off"
                   :
                   : "v"(g), "v"(l)
                   : "memory");
    }
    asm volatile("s_wait_asynccnt 0x0" ::: "memory");
  }
}

extern "C" void kernel_launch(void* const* d_in, const int* in_sizes, int n_in,
                              void* d_out, int out_size, void* d_ws,
                              size_t ws_size, hipStream_t stream) {
  const float* x           = (const float*)d_in[0];
  const float* rel_loc     = (const float*)d_in[1];
  const float* rel_rot_ref = (const float*)d_in[2];
  float* out               = (float*)d_out;
  float* cst               = (float*)d_ws; // 26*8 floats

  const int batch  = in_sizes[0] / ELEM_FLOATS; // 131072
  const int blocks = batch / TPB;               // 1024

  fk_setup<<<1, 32, 0, stream>>>(rel_rot_ref, rel_loc, cst);
  fk_kernel<<<blocks, TPB, 0, stream>>>(x, cst, out);
}